// Criterion_188978561412
// MI455X (gfx1250) — compile-verified
//
#include <hip/hip_runtime.h>
#include <math.h>

typedef __attribute__((ext_vector_type(2))) float v2f;
typedef __attribute__((ext_vector_type(8))) float v8f;

#define MARGIN_F 0.2f
#define EPS_F 1e-8f
#define DIM 512
#define NBLOCKS 1024
#define WAVES_PER_BLOCK 8

// ---------------------------------------------------------------------------
// Stage 1: one wave per triplet (grid-stride). D=512 floats = 32 lanes x 4
// float4 -> fully coalesced global_load_b128. Diff^2 accumulated per lane,
// 5-step shfl_xor cross-lane reduction, per-wave accumulators, deterministic
// in-order per-block combine (no float atomics -> bitwise-stable replays).
// Prefetch next triplet's 3 rows (32 lanes x 64B = 2KB = one full row each)
// to overlap L2 gather latency with compute.
// ---------------------------------------------------------------------------
__global__ __launch_bounds__(256) void triplet_partial(
    const float* __restrict__ batch, const float* __restrict__ beta,
    const int* __restrict__ labels, const int* __restrict__ triplets,
    float* __restrict__ totals, float* __restrict__ counts, int T)
{
    __shared__ float wsum[WAVES_PER_BLOCK];
    __shared__ float wcnt[WAVES_PER_BLOCK];

    const int lane  = threadIdx.x & 31;
    const int wave  = threadIdx.x >> 5;
    const int gwave = blockIdx.x * WAVES_PER_BLOCK + wave;
    const int nwave = gridDim.x * WAVES_PER_BLOCK;

    float acc_total = 0.0f;
    float acc_count = 0.0f;

    for (int t = gwave; t < T; t += nwave) {
        const int ia = triplets[3 * t + 0];
        const int ip = triplets[3 * t + 1];
        const int in = triplets[3 * t + 2];

        // Prefetch next iteration's rows while we compute this one.
        const int t2 = t + nwave;
        if (t2 < T) {
            const int ja = triplets[3 * t2 + 0];
            const int jp = triplets[3 * t2 + 1];
            const int jn = triplets[3 * t2 + 2];
            const char* pa = (const char*)(batch + (size_t)ja * DIM) + lane * 64;
            const char* pp = (const char*)(batch + (size_t)jp * DIM) + lane * 64;
            const char* pn = (const char*)(batch + (size_t)jn * DIM) + lane * 64;
            __builtin_prefetch(pa, 0, 3);
            __builtin_prefetch(pp, 0, 3);
            __builtin_prefetch(pn, 0, 3);
        }

        const float4* A = (const float4*)(batch + (size_t)ia * DIM);
        const float4* P = (const float4*)(batch + (size_t)ip * DIM);
        const float4* N = (const float4*)(batch + (size_t)in * DIM);

        float s_ap = 0.0f, s_an = 0.0f;
#pragma unroll
        for (int j = 0; j < 4; ++j) {
            const int idx = lane + j * 32;   // 128 float4 per row, coalesced
            float4 av = A[idx];
            float4 pv = P[idx];
            float4 nv = N[idx];
            float dx = av.x - pv.x, dy = av.y - pv.y,
                  dz = av.z - pv.z, dw = av.w - pv.w;
            s_ap += dx * dx + dy * dy + dz * dz + dw * dw;
            dx = av.x - nv.x; dy = av.y - nv.y;
            dz = av.z - nv.z; dw = av.w - nv.w;
            s_an += dx * dx + dy * dy + dz * dz + dw * dw;
        }

        // wave32 cross-lane tree reduction
#pragma unroll
        for (int off = 16; off > 0; off >>= 1) {
            s_ap += __shfl_xor(s_ap, off, 32);
            s_an += __shfl_xor(s_an, off, 32);
        }

        if (lane == 0) {
            const float d_ap = sqrtf(s_ap + EPS_F);
            const float d_an = sqrtf(s_an + EPS_F);
            const float b    = beta[labels[ia]];
            const float pos  = fmaxf(d_ap - b + MARGIN_F, 0.0f);
            const float neg  = fmaxf(b - d_an + MARGIN_F, 0.0f);
            acc_total += pos + neg;
            acc_count += (pos > 0.0f || neg > 0.0f) ? 1.0f : 0.0f;
        }
    }

    if (lane == 0) {
        wsum[wave] = acc_total;
        wcnt[wave] = acc_count;
    }
    __syncthreads();

    if (threadIdx.x == 0) {
        float ts = 0.0f, cs = 0.0f;
#pragma unroll
        for (int w = 0; w < WAVES_PER_BLOCK; ++w) { ts += wsum[w]; cs += wcnt[w]; }
        totals[blockIdx.x] = ts;
        counts[blockIdx.x] = cs;
    }
}

// ---------------------------------------------------------------------------
// Stage 2: single wave. Reduce the 1024 (total,count) partials with
// V_WMMA_F32_16X16X4_F32:  C += A(16x4 data) x B(4x16 all-ones)
// => every column of C accumulates the row-sums of A; one WMMA folds 64
// partials. B = all-ones is layout-invariant; A layout per ISA 7.12.2
// (lanes 0-15: K=0,1 ; lanes 16-31: K=2,3). C column 0 lives in lane 0
// (M=0..7 across the 8 C VGPRs) and lane 16 (M=8..15), so a local 8-sum
// plus one shfl_xor(16) yields the grand total.
// EXEC is all-ones throughout (uniform single-wave control flow).
// ---------------------------------------------------------------------------
__global__ __launch_bounds__(32) void final_reduce(
    const float* __restrict__ totals, const float* __restrict__ counts,
    float* __restrict__ out, int nparts)
{
    const int lane = threadIdx.x;  // 0..31, all active

    v2f ones; ones.x = 1.0f; ones.y = 1.0f;
    v8f ct = {};
    v8f cc = {};

    // A[m][k] = partials[64*i + 4*m + k]
    const int base = (lane < 16) ? (4 * lane) : (4 * (lane - 16) + 2);

    for (int i = 0; i < nparts; i += 64) {
        v2f a;
        a.x = totals[i + base + 0];
        a.y = totals[i + base + 1];
        ct = __builtin_amdgcn_wmma_f32_16x16x4_f32(
                false, a, false, ones, (short)0, ct, false, false);
        a.x = counts[i + base + 0];
        a.y = counts[i + base + 1];
        cc = __builtin_amdgcn_wmma_f32_16x16x4_f32(
                false, a, false, ones, (short)0, cc, false, false);
    }

    // Sum C column for this lane's half (M=0..7 here, M=8..15 in lane^16).
    float myt = ct[0] + ct[1] + ct[2] + ct[3] + ct[4] + ct[5] + ct[6] + ct[7];
    float myc = cc[0] + cc[1] + cc[2] + cc[3] + cc[4] + cc[5] + cc[6] + cc[7];
    myt += __shfl_xor(myt, 16, 32);
    myc += __shfl_xor(myc, 16, 32);

    if (lane == 0) {
        const float loss = (myc > 0.0f) ? (myt / fmaxf(myc, 1.0f)) : myt;
        out[0] = loss;
    }
}

extern "C" void kernel_launch(void* const* d_in, const int* in_sizes, int n_in,
                              void* d_out, int out_size, void* d_ws, size_t ws_size,
                              hipStream_t stream) {
    const float* batch    = (const float*)d_in[0];  // (4096, 512) f32
    const float* beta     = (const float*)d_in[1];  // (100,) f32
    const int*   labels   = (const int*)d_in[2];    // (4096,) i32
    const int*   triplets = (const int*)d_in[3];    // (65536, 3) i32
    float*       out      = (float*)d_out;

    const int T = in_sizes[3] / 3;

    float* totals = (float*)d_ws;        // NBLOCKS floats
    float* counts = totals + NBLOCKS;    // NBLOCKS floats

    triplet_partial<<<NBLOCKS, 256, 0, stream>>>(
        batch, beta, labels, triplets, totals, counts, T);
    final_reduce<<<1, 32, 0, stream>>>(totals, counts, out, NBLOCKS);
}